// GIN_50835232915932
// MI455X (gfx1250) — compile-verified
//
#include <hip/hip_runtime.h>

typedef float v2f __attribute__((ext_vector_type(2)));
typedef float v8f __attribute__((ext_vector_type(8)));

#define C_DIM 128
#define BN_EPS 1e-5f

__device__ __forceinline__ void atomAddF(float* p, float v) {
    // unsafeAtomicAdd forces the native global_atomic_add_f32 path on AMD
    // (no CAS loop); pooled/agg are coarse-grained device memory, so safe.
    unsafeAtomicAdd(p, v);
}

// ---------------------------------------------------------------------------
// Edge scatter: agg[dst] += h[src].  One wave per edge, 4 channels per lane.
// Gathers are 512B contiguous per wave; atomics are lane-contiguous.
// Whole working set (h, agg: 51.2MB each) is L2-resident (192MB).
// ---------------------------------------------------------------------------
__global__ __launch_bounds__(256)
void gin_scatter(const float* __restrict__ h, const int* __restrict__ edges,
                 long E, float* __restrict__ agg) {
    long gid = (long)blockIdx.x * blockDim.x + threadIdx.x;
    long e = gid >> 5;
    if (e >= E) return;
    int cg = (int)(gid & 31) * 4;
    long src = edges[e];
    long dst = edges[E + e];
    float4 v = *(const float4*)(h + src * C_DIM + cg);
    float* a = agg + dst * C_DIM + cg;
    atomAddF(a + 0, v.x);
    atomAddF(a + 1, v.y);
    atomAddF(a + 2, v.z);
    atomAddF(a + 3, v.w);
}

// ---------------------------------------------------------------------------
// GIN dense layer via fp32 WMMA: z = (h+agg) @ W + b; BN; ReLU.
// One block per 16-row tile; wave w computes column tile [16w, 16w+16).
// Epilogue fuses global_add_pool into `pooled` via atomics.
// ---------------------------------------------------------------------------
#if defined(__HIP_DEVICE_COMPILE__)
#if !__has_builtin(__builtin_amdgcn_wmma_f32_16x16x4_f32)
#error "gfx1250 f32 WMMA builtin __builtin_amdgcn_wmma_f32_16x16x4_f32 not available"
#endif
#endif

#define HT_STRIDE 132  // pad 128 -> 132 floats: LDS banks = k0 + 4*ln (conflict-free)

__global__ __launch_bounds__(256)
void gin_gemm(const float* __restrict__ h, const float* __restrict__ agg,
              const float* __restrict__ W, const float* __restrict__ bias,
              const float* __restrict__ gam, const float* __restrict__ bet,
              const float* __restrict__ mu, const float* __restrict__ var,
              const int* __restrict__ batch, float* __restrict__ hout,
              float* __restrict__ pooled, int colOfs) {
    __shared__ float Ht[16 * HT_STRIDE];

    const int tile = blockIdx.x;
    const long base = (long)tile * 16 * C_DIM;

    // Stage (h + agg) 16x128 tile into LDS (contiguous 8KB read).
    for (int i = threadIdx.x; i < 16 * C_DIM; i += 256) {
        int r = i >> 7, c = i & 127;
        Ht[r * HT_STRIDE + c] = h[base + i] + agg[base + i];
    }
    __syncthreads();

    const int wave = threadIdx.x >> 5;   // column tile id (0..7)
    const int lane = threadIdx.x & 31;
    const int hi = lane >> 4;            // 0: K lanes {0,1}, 1: K lanes {2,3}
    const int ln = lane & 15;
    const int n0 = wave * 16;

    v8f acc = {};
    #pragma unroll
    for (int kb = 0; kb < C_DIM; kb += 4) {
        const int k0 = kb + 2 * hi;
        v2f a, b;
        // A 16x4 layout: lanes 0-15 M=ln K={kb,kb+1}; lanes 16-31 M=ln K={kb+2,kb+3}
        a.x = Ht[ln * HT_STRIDE + k0];
        a.y = Ht[ln * HT_STRIDE + k0 + 1];
        // B 4x16 layout: lane -> column N=n0+ln; VGPR0/1 -> K pair per lane half
        b.x = W[k0 * C_DIM + n0 + ln];
        b.y = W[(k0 + 1) * C_DIM + n0 + ln];
        acc = __builtin_amdgcn_wmma_f32_16x16x4_f32(
            false, a, false, b, (short)0, acc, false, false);
    }

    // Epilogue: D layout -> VGPR r holds (M = r + 8*hi, N = n0+ln).
    const int col = n0 + ln;
    const float bb = bias[col];
    const float g  = gam[col];
    const float bt = bet[col];
    const float m  = mu[col];
    const float rs = rsqrtf(var[col] + BN_EPS);
    #pragma unroll
    for (int r = 0; r < 8; ++r) {
        const int row = tile * 16 + r + 8 * hi;
        float z = acc[r] + bb;
        z = (z - m) * rs * g + bt;
        z = fmaxf(z, 0.0f);
        if (hout) hout[(long)row * C_DIM + col] = z;
        const int gidx = batch[row];
        atomAddF(&pooled[gidx * 512 + colOfs + col], z);
    }
}

// ---------------------------------------------------------------------------
// Final MLP: relu(pooled @ W1 + b1) @ W2 + b2, plus log_softmax. 1 block/graph.
// ---------------------------------------------------------------------------
__global__ __launch_bounds__(256)
void gin_mlp(const float* __restrict__ pooled,
             const float* __restrict__ W1, const float* __restrict__ b1,
             const float* __restrict__ W2, const float* __restrict__ b2,
             float* __restrict__ out, int numGraphs) {
    __shared__ float p[512];
    __shared__ float hid[256];
    __shared__ float o10[10];
    __shared__ float red[2];
    const int g = blockIdx.x, t = threadIdx.x;

    for (int i = t; i < 512; i += 256) p[i] = pooled[g * 512 + i];
    __syncthreads();

    float s = b1[t];
    for (int k = 0; k < 512; ++k) s += p[k] * W1[k * 256 + t];
    hid[t] = fmaxf(s, 0.0f);
    __syncthreads();

    if (t < 10) {
        float s2 = b2[t];
        for (int j = 0; j < 256; ++j) s2 += hid[j] * W2[j * 10 + t];
        o10[t] = s2;
        out[g * 10 + t] = s2;                       // output 0: logits
    }
    __syncthreads();

    if (t == 0) {
        float mx = o10[0];
        for (int i = 1; i < 10; ++i) mx = fmaxf(mx, o10[i]);
        float se = 0.0f;
        for (int i = 0; i < 10; ++i) se += expf(o10[i] - mx);
        red[0] = mx;
        red[1] = logf(se);
    }
    __syncthreads();

    if (t < 10)                                     // output 1: log_softmax
        out[numGraphs * 10 + g * 10 + t] = o10[t] - red[0] - red[1];
}

// ---------------------------------------------------------------------------
extern "C" void kernel_launch(void* const* d_in, const int* in_sizes, int n_in,
                              void* d_out, int out_size, void* d_ws, size_t ws_size,
                              hipStream_t stream) {
    const float* x     = (const float*)d_in[0];
    const int*   ex    = (const int*)  d_in[1];
    const int*   ey    = (const int*)  d_in[2];
    const int*   batch = (const int*)  d_in[3];
    const float* convW = (const float*)d_in[4];
    const float* convB = (const float*)d_in[5];
    const float* bnG   = (const float*)d_in[6];
    const float* bnB   = (const float*)d_in[7];
    const float* bnM   = (const float*)d_in[8];
    const float* bnV   = (const float*)d_in[9];
    const float* W1    = (const float*)d_in[10];
    const float* b1    = (const float*)d_in[11];
    const float* W2    = (const float*)d_in[12];
    const float* b2    = (const float*)d_in[13];

    const long N = in_sizes[0] / C_DIM;   // 100000
    const long E = in_sizes[1] / 2;       // 1600000
    const int  numGraphs = 128;

    float* agg    = (float*)d_ws;                 // N*128 f32
    float* h1     = agg + N * C_DIM;              // N*128 f32
    float* pooled = h1 + N * C_DIM;               // 128*512 f32

    (void)hipMemsetAsync(pooled, 0, (size_t)numGraphs * 512 * sizeof(float), stream);

    const long scatterThreads = E * 32;
    const int  scatterBlocks  = (int)((scatterThreads + 255) / 256);
    const int  gemmBlocks     = (int)(N / 16);

    auto conv = [&](const float* hin, const int* edges, int idx,
                    float* hStore, int colOfs) {
        (void)hipMemsetAsync(agg, 0, (size_t)N * C_DIM * sizeof(float), stream);
        gin_scatter<<<scatterBlocks, 256, 0, stream>>>(hin, edges, E, agg);
        gin_gemm<<<gemmBlocks, 256, 0, stream>>>(
            hin, agg,
            convW + (long)idx * C_DIM * C_DIM, convB + idx * C_DIM,
            bnG + idx * C_DIM, bnB + idx * C_DIM,
            bnM + idx * C_DIM, bnV + idx * C_DIM,
            batch, hStore, pooled, colOfs);
    };

    conv(x,  ex, 0, h1,      0);     // h1_x  (stored, pooled)
    conv(h1, ex, 1, nullptr, 128);   // h2_x  (pool only)
    conv(x,  ey, 2, h1,      256);   // h1_y  (stored, pooled; reuses h1)
    conv(h1, ey, 3, nullptr, 384);   // h2_y  (pool only)

    gin_mlp<<<numGraphs, 256, 0, stream>>>(pooled, W1, b1, W2, b2,
                                           (float*)d_out, numGraphs);
}